// LRU_88845693485287
// MI455X (gfx1250) — compile-verified
//
#include <hip/hip_runtime.h>

typedef __attribute__((ext_vector_type(2))) float v2f;
typedef __attribute__((ext_vector_type(4))) float v4f;
typedef __attribute__((ext_vector_type(8))) float v8f;

#define B_    16
#define L_    4096
#define H_    128
#define HOUT_ 128
#define N_    256
#define NCH_  64              // chunks per sequence for blocked scan
#define CLEN_ (L_ / NCH_)     // 64 steps per chunk

// D = A(16x4 f32) * B(4x16 f32) + C(16x16 f32), full fp32 matrix pipe.
static __device__ __forceinline__ v8f wmma_f32(v2f a, v2f b, v8f c) {
  return __builtin_amdgcn_wmma_f32_16x16x4_f32(false, a, false, b, (short)0, c,
                                               false, false);
}

// ---------------------------------------------------------------------------
// Stage 1: Bu[b,l,n] = sum_h input[b,l,h] * g[n]*Bp{re,im}[n,h]
// Written interleaved (re,im) into states slot (b, l+1, n).
// One wave computes a 32(M)x16(n) tile (2 row-tiles x {re,im} = 4 accums),
// so each scaled Bp fragment feeds two WMMAs.
// ---------------------------------------------------------------------------
__global__ void __launch_bounds__(256)
lru_bu(const float* __restrict__ in, const float* __restrict__ Bp_re,
       const float* __restrict__ Bp_im, const float* __restrict__ gamma_log,
       float* __restrict__ states) {
  const int lane  = threadIdx.x & 31;
  const int tile  = blockIdx.x * 8 + (threadIdx.x >> 5);
  const int m0    = (tile >> 4) << 5;        // 32-row block over B*L
  const int n0    = (tile & 15) << 4;        // 16-col block over N
  const int sub   = lane & 15;
  const int khalf = (lane >> 4) << 1;        // lanes 16..31 carry K+2,K+3
  const int col   = n0 + sub;

  const float g = __expf(gamma_log[col]);
  const float* arow0 = in    + (size_t)(m0 + sub) * H_ + khalf;
  const float* arow1 = arow0 + (size_t)16 * H_;
  const float* brow  = Bp_re + (size_t)col * H_ + khalf;
  const float* irow  = Bp_im + (size_t)col * H_ + khalf;

  v8f ar0 = {0.f,0.f,0.f,0.f,0.f,0.f,0.f,0.f};
  v8f ai0 = ar0, ar1 = ar0, ai1 = ar0;
#pragma unroll 4
  for (int k = 0; k < H_; k += 4) {
    v2f a0 = *(const v2f*)(arow0 + k);
    v2f a1 = *(const v2f*)(arow1 + k);
    v2f br = *(const v2f*)(brow + k) * g;
    v2f bi = *(const v2f*)(irow + k) * g;
    ar0 = wmma_f32(a0, br, ar0);
    ai0 = wmma_f32(a0, bi, ai0);
    ar1 = wmma_f32(a1, br, ar1);
    ai1 = wmma_f32(a1, bi, ai1);
  }

  // C/D layout: VGPR i -> rows m+i (lanes 0..15) and m+8+i (lanes 16..31).
  const int mbase = m0 + ((lane >> 4) << 3);
#pragma unroll
  for (int i = 0; i < 8; ++i) {
    const int r0 = mbase + i;                      // flattened (b,l), L=4096
    const int r1 = r0 + 16;
    const size_t s0 = (size_t)(r0 + (r0 >> 12) + 1) * (2 * N_); // (b, l+1)
    const size_t s1 = (size_t)(r1 + (r1 >> 12) + 1) * (2 * N_);
    v2f v0 = {ar0[i], ai0[i]};
    v2f v1 = {ar1[i], ai1[i]};
    *(v2f*)(states + s0 + 2 * col) = v0;
    *(v2f*)(states + s1 + 2 * col) = v1;
  }
}

// ---------------------------------------------------------------------------
// Stage 2a: per-(b, chunk, n) carry-only scan from zero init (read-only over
// Bu, no write-back).  By linearity: true_final = local_final + lam^C * s_in.
// ---------------------------------------------------------------------------
__global__ void __launch_bounds__(256)
lru_carry_local(const float* __restrict__ nu_log,
                const float* __restrict__ theta_log,
                const float* __restrict__ states,
                float* __restrict__ carry_out) {
  const int tid = blockIdx.x * blockDim.x + threadIdx.x;   // B*NCH*N
  const int n = tid % N_;
  const int c = (tid / N_) % NCH_;
  const int b = tid / (N_ * NCH_);

  const float mag = __expf(-__expf(nu_log[n]));
  const float th  = __expf(theta_log[n]);
  const float lr  = mag * __cosf(th);
  const float li  = mag * __sinf(th);

  float xr = 0.f, xi = 0.f;
  const float* base = states +
      ((size_t)b * (L_ + 1) + (size_t)c * CLEN_ + 1) * (2 * N_) + 2 * n;
#pragma unroll 4
  for (int j = 0; j < CLEN_; ++j) {
    const float* p = base + (size_t)j * (2 * N_);
    const float br = p[0], bi = p[1];
    const float nr = lr * xr - li * xi + br;
    const float ni = lr * xi + li * xr + bi;
    xr = nr; xi = ni;
  }
  float* co = carry_out + ((size_t)(b * NCH_ + c) * N_ + n) * 2;
  co[0] = xr; co[1] = xi;
}

// ---------------------------------------------------------------------------
// Stage 2b: combine chunk carries with lam^CLEN (closed form):
//   s_0 = x0 ; s_{c+1} = lam^C * s_c + local_c ; emit s_c per chunk.
// Also writes states[b,0,n] = x0.
// ---------------------------------------------------------------------------
__global__ void __launch_bounds__(256)
lru_carry_combine(const float* __restrict__ state0,
                  const float* __restrict__ nu_log,
                  const float* __restrict__ theta_log,
                  const float* __restrict__ carry_local,
                  float* __restrict__ carry_in,
                  float* __restrict__ states) {
  const int tid = blockIdx.x * blockDim.x + threadIdx.x;   // B*N
  const int n = tid % N_;
  const int b = tid / N_;

  const float en = __expf(nu_log[n]);
  const float th = __expf(theta_log[n]);
  const float magC = __expf(-(float)CLEN_ * en);     // |lam|^CLEN
  const float angC = (float)CLEN_ * th;
  const float cr = magC * __cosf(angC);
  const float ci = magC * __sinf(angC);

  float sr = state0[b * N_ + n], si = 0.f;           // incoming state = x0
  for (int c = 0; c < NCH_; ++c) {
    const size_t off = ((size_t)(b * NCH_ + c) * N_ + n) * 2;
    carry_in[off] = sr; carry_in[off + 1] = si;
    const float fr = carry_local[off], fi = carry_local[off + 1];
    const float nr = cr * sr - ci * si + fr;
    const float ni = cr * si + ci * sr + fi;
    sr = nr; si = ni;
  }
  float* s0p = states + (size_t)b * (L_ + 1) * (2 * N_) + 2 * n;
  s0p[0] = state0[b * N_ + n];
  s0p[1] = 0.f;
}

// ---------------------------------------------------------------------------
// Stage 2c: real scan per chunk starting from the correct incoming carry;
// overwrites Bu slots with the true states (single read+write pass).
// ---------------------------------------------------------------------------
__global__ void __launch_bounds__(256)
lru_scan_write(const float* __restrict__ nu_log,
               const float* __restrict__ theta_log,
               const float* __restrict__ carry_in,
               float* __restrict__ states) {
  const int tid = blockIdx.x * blockDim.x + threadIdx.x;   // B*NCH*N
  const int n = tid % N_;
  const int c = (tid / N_) % NCH_;
  const int b = tid / (N_ * NCH_);

  const float mag = __expf(-__expf(nu_log[n]));
  const float th  = __expf(theta_log[n]);
  const float lr  = mag * __cosf(th);
  const float li  = mag * __sinf(th);

  const size_t coff = ((size_t)(b * NCH_ + c) * N_ + n) * 2;
  float xr = carry_in[coff], xi = carry_in[coff + 1];

  float* base = states +
      ((size_t)b * (L_ + 1) + (size_t)c * CLEN_ + 1) * (2 * N_) + 2 * n;
#pragma unroll 4
  for (int j = 0; j < CLEN_; ++j) {
    float* p = base + (size_t)j * (2 * N_);
    const float br = p[0], bi = p[1];
    const float nr = lr * xr - li * xi + br;
    const float ni = lr * xi + li * xr + bi;
    xr = nr; xi = ni;
    p[0] = xr; p[1] = xi;
  }
}

// ---------------------------------------------------------------------------
// Stage 3: out[b,l,o] = Re(C * x_pre) + D u
//        = C_re.x_re - C_im.x_im (K=256) + D.u (K=128), all via fp32 WMMA.
// One wave per 32(M)x16(o) tile: each C_re/C_im/D fragment feeds 2 WMMAs.
// ---------------------------------------------------------------------------
__global__ void __launch_bounds__(256)
lru_out(const float* __restrict__ in, const float* __restrict__ C_re,
        const float* __restrict__ C_im, const float* __restrict__ Dm,
        const float* __restrict__ states, float* __restrict__ out) {
  const int lane  = threadIdx.x & 31;
  const int tile  = blockIdx.x * 8 + (threadIdx.x >> 5);
  const int m0    = (tile >> 3) << 5;          // 32-row block over B*L
  const int o0    = (tile & 7) << 4;           // 16-col block over HOUT
  const int sub   = lane & 15;
  const int khalf = (lane >> 4) << 1;
  const int row0  = m0 + sub;
  const int row1  = row0 + 16;
  const int o     = o0 + sub;

  const float* arow0 = in + (size_t)row0 * H_ + khalf;
  const float* arow1 = in + (size_t)row1 * H_ + khalf;
  const float* drow  = Dm + (size_t)o * H_ + khalf;
  const float* crrow = C_re + (size_t)o * N_ + khalf;
  const float* cirow = C_im + (size_t)o * N_ + khalf;
  // pre-update state x_l lives in slot (b, l) = row + row/L
  const float* srow0 =
      states + (size_t)(row0 + (row0 >> 12)) * (2 * N_) + 2 * khalf;
  const float* srow1 =
      states + (size_t)(row1 + (row1 >> 12)) * (2 * N_) + 2 * khalf;

  v8f acc0 = {0.f,0.f,0.f,0.f,0.f,0.f,0.f,0.f};
  v8f acc1 = acc0;

  // D @ u term, K = H = 128
#pragma unroll 8
  for (int k = 0; k < H_; k += 4) {
    v2f a0 = *(const v2f*)(arow0 + k);
    v2f a1 = *(const v2f*)(arow1 + k);
    v2f bd = *(const v2f*)(drow + k);
    acc0 = wmma_f32(a0, bd, acc0);
    acc1 = wmma_f32(a1, bd, acc1);
  }
  // Re(C x) = C_re.x_re - C_im.x_im, K = N = 256; one 16B load per row feeds
  // both the C_re and (negated) C_im fragments.
#pragma unroll 4
  for (int k = 0; k < N_; k += 4) {
    v4f s40 = *(const v4f*)(srow0 + 2 * k);    // {re,im,re,im}
    v4f s41 = *(const v4f*)(srow1 + 2 * k);
    v2f are0 = { s40.x,  s40.z};
    v2f aim0 = {-s40.y, -s40.w};
    v2f are1 = { s41.x,  s41.z};
    v2f aim1 = {-s41.y, -s41.w};
    v2f bre = *(const v2f*)(crrow + k);
    v2f bim = *(const v2f*)(cirow + k);
    acc0 = wmma_f32(are0, bre, acc0);
    acc0 = wmma_f32(aim0, bim, acc0);
    acc1 = wmma_f32(are1, bre, acc1);
    acc1 = wmma_f32(aim1, bim, acc1);
  }

  const int mbase = m0 + ((lane >> 4) << 3);
#pragma unroll
  for (int i = 0; i < 8; ++i) {
    out[(size_t)(mbase + i) * HOUT_ + o]      = acc0[i];
    out[(size_t)(mbase + 16 + i) * HOUT_ + o] = acc1[i];
  }
}

// ---------------------------------------------------------------------------
extern "C" void kernel_launch(void* const* d_in, const int* in_sizes, int n_in,
                              void* d_out, int out_size, void* d_ws, size_t ws_size,
                              hipStream_t stream) {
  (void)in_sizes; (void)n_in; (void)out_size; (void)ws_size;
  const float* input     = (const float*)d_in[0];
  const float* state0    = (const float*)d_in[1];
  const float* nu_log    = (const float*)d_in[2];
  const float* theta_log = (const float*)d_in[3];
  const float* gamma_log = (const float*)d_in[4];
  const float* Bp_re     = (const float*)d_in[5];
  const float* Bp_im     = (const float*)d_in[6];
  const float* C_re      = (const float*)d_in[7];
  const float* C_im      = (const float*)d_in[8];
  const float* Dm        = (const float*)d_in[9];

  float* out    = (float*)d_out;                         // (B, L, HOUT)
  float* states = out + (size_t)B_ * L_ * HOUT_;         // (B, L+1, N) cplx

  float* carry_local = (float*)d_ws;                             // B*NCH*N*2
  float* carry_in    = carry_local + (size_t)B_ * NCH_ * N_ * 2; // B*NCH*N*2

  const dim3 blk(256);
  lru_bu<<<(B_ * L_ / 32) * (N_ / 16) / 8, blk, 0, stream>>>(
      input, Bp_re, Bp_im, gamma_log, states);
  lru_carry_local<<<B_ * NCH_ * N_ / 256, blk, 0, stream>>>(
      nu_log, theta_log, states, carry_local);
  lru_carry_combine<<<B_ * N_ / 256, blk, 0, stream>>>(
      state0, nu_log, theta_log, carry_local, carry_in, states);
  lru_scan_write<<<B_ * NCH_ * N_ / 256, blk, 0, stream>>>(
      nu_log, theta_log, carry_in, states);
  lru_out<<<(B_ * L_ / 32) * (HOUT_ / 16) / 8, blk, 0, stream>>>(
      input, C_re, C_im, Dm, states, out);
}